// TransformerCRF_42545946034328
// MI455X (gfx1250) — compile-verified
//
#include <hip/hip_runtime.h>
#include <hip/hip_bf16.h>

typedef float v2f __attribute__((ext_vector_type(2)));
typedef float v8f __attribute__((ext_vector_type(8)));

#define KC 32
#define A_STRIDE 36   // 32 + 4 pad: row*36 mod 64 hits 16 distinct banks
#define NCH (768 / KC)

// wait until this wave's outstanding async(global<->LDS) ops <= n
#define WAIT_ASYNC(n) asm volatile("s_wait_asynccnt %0" :: "i"(n) : "memory")

// ---------------------------------------------------------------------------
// Kernel 1: logits = hidden @ W + b   via V_WMMA_F32_16X16X4_F32 with
// double-buffered GLOBAL_LOAD_ASYNC_TO_LDS_B128 staging of the A tiles.
// grid 256 blocks x 256 threads; each block: 128 rows, each wave: 16 rows.
// ---------------------------------------------------------------------------
__global__ __launch_bounds__(256) void logits_wmma_kernel(
    const float* __restrict__ hidden,   // (32768, 768)
    const float* __restrict__ W,        // (768, 9)
    const float* __restrict__ bvec,     // (9)
    float* __restrict__ logits)         // (32768, 9)
{
    __shared__ float Alds[2][128 * A_STRIDE];   // 36 KB
    __shared__ float Wlds[2][KC * 16];          // 4 KB (L padded 9->16)

    const int tid  = threadIdx.x;
    const int wave = tid >> 5;
    const int lane = tid & 31;
    const int m     = lane & 15;            // A row within tile / B,C column
    const int khalf = (lane >> 4) * 2;      // 0 for lanes 0-15, 2 for 16-31
    const long row_base = (long)blockIdx.x * 128;

    // LDS byte offsets of the two A buffers (generic ptr low 32 bits = LDS off)
    const unsigned abase0 = (unsigned)(unsigned long long)(const void*)&Alds[0][0];
    const unsigned abase1 = (unsigned)(unsigned long long)(const void*)&Alds[1][0];

    // per-thread fixed part of the async copy addressing (4 b128 per thread)
    const int r_cp  = (tid * 4) >> 5;                 // reused below via idx math

    v8f c = {};

    // ---- stage-issue helpers (inlined manually via macros) ----
#define ISSUE_A(c_idx, buf)                                                   \
    do {                                                                      \
        const unsigned ab = (buf) ? abase1 : abase0;                          \
        const int k0_ = (c_idx) * KC;                                         \
        _Pragma("unroll")                                                     \
        for (int i_ = 0; i_ < 4; ++i_) {                                      \
            const int idx_ = tid + i_ * 256;       /* 0..1023 */              \
            const int r_   = idx_ >> 3;            /* 0..127  */              \
            const int kq_  = (idx_ & 7) * 4;       /* 0..28   */              \
            const float* g_ = &hidden[(row_base + r_) * 768 + k0_ + kq_];     \
            const unsigned loff_ = ab + (unsigned)((r_ * A_STRIDE + kq_) * 4);\
            asm volatile("global_load_async_to_lds_b128 %0, %1, off"          \
                         :: "v"(loff_), "v"(g_) : "memory");                  \
        }                                                                     \
    } while (0)

#define ISSUE_W(c_idx, buf)                                                   \
    do {                                                                      \
        const int k0_ = (c_idx) * KC;                                         \
        _Pragma("unroll")                                                     \
        for (int i_ = 0; i_ < 2; ++i_) {                                      \
            const int idx_ = tid + i_ * 256;       /* 0..511 */               \
            const int k_ = idx_ >> 4, l_ = idx_ & 15;                         \
            Wlds[buf][idx_] = (l_ < 9) ? W[(k0_ + k_) * 9 + l_] : 0.0f;       \
        }                                                                     \
    } while (0)

    // prologue: chunk 0 into buffer 0
    ISSUE_A(0, 0);
    ISSUE_W(0, 0);

    for (int ch = 0; ch < NCH; ++ch) {
        const int p = ch & 1;
        if (ch + 1 < NCH) {
            ISSUE_A(ch + 1, 1 - p);   // overlap next chunk's copy with compute
            ISSUE_W(ch + 1, 1 - p);
            WAIT_ASYNC(4);            // chunk ch's 4 async ops (per wave) done
        } else {
            WAIT_ASYNC(0);
        }
        __syncthreads();              // chunk ch visible to all waves

        const float* arow = &Alds[p][(wave * 16 + m) * A_STRIDE];
        const float* wbuf = &Wlds[p][0];
#pragma unroll
        for (int kk = 0; kk < KC; kk += 4) {
            v2f a, bf;
            a[0] = arow[kk + khalf];
            a[1] = arow[kk + khalf + 1];
            const float* wp = &wbuf[(kk + khalf) * 16 + m];
            bf[0] = wp[0];
            bf[1] = wp[16];
            c = __builtin_amdgcn_wmma_f32_16x16x4_f32(
                    false, a, false, bf, (short)0, c, false, false);
        }
        __syncthreads();              // all waves done with buffer p
    }
#undef ISSUE_A
#undef ISSUE_W
    (void)r_cp;

    // epilogue: C layout — VGPR i holds M = i + (lane>=16 ? 8 : 0), N = lane&15
    const int col = m;
    if (col < 9) {
        const float bias = bvec[col];
        const long r0 = row_base + wave * 16 + ((lane >> 4) ? 8 : 0);
#pragma unroll
        for (int i = 0; i < 8; ++i)
            logits[(r0 + i) * 9 + col] = c[i] + bias;
    }
}

// ---------------------------------------------------------------------------
// Kernel 2: per-batch CRF forward (log-norm), numerator, Viterbi + backtrack.
// grid 64 blocks x 32 threads (one wave per batch); lane j owns state j.
// ---------------------------------------------------------------------------
__global__ __launch_bounds__(32) void crf_kernel(
    const float* __restrict__ logits,       // (64*512, 9)
    const int* __restrict__ mask,           // (64, 512)
    const int* __restrict__ labels,         // (64, 512)
    const float* __restrict__ start_t,      // (9)
    const float* __restrict__ end_t,        // (9)
    const float* __restrict__ trans,        // (9, 9)
    float* __restrict__ numden,             // ws: [0..63]=num, [64..127]=den
    float* __restrict__ out_fake)           // (64, 512, 9)
{
    __shared__ unsigned char bp_lds[512 * 9];
    __shared__ unsigned char mask_lds[512];
    __shared__ short tags_lds[512];

    const int b = blockIdx.x;
    const int j = threadIdx.x;
    const bool valid = (j < 9);
    const float NEG = -1e30f;
    const long base = (long)b * 512;

    float trcol[9];
#pragma unroll
    for (int i = 0; i < 9; ++i) trcol[i] = valid ? trans[i * 9 + j] : 0.0f;

    float s = valid ? (start_t[j] + logits[base * 9 + j]) : NEG;
    float v = s;

    // lane-0 numerator state
    const int m0 = mask[base];
    int lab0 = labels[base];
    int tag_prev = (lab0 == -100) ? 0 : lab0;
    float num = start_t[tag_prev] + logits[base * 9 + tag_prev];
    int len = m0;
    if (j == 0) mask_lds[0] = (unsigned char)m0;

    for (int t = 1; t < 512; ++t) {
        const int mt = mask[base + t];
        const float em = valid ? logits[(base + t) * 9 + j] : NEG;

        float si[9], vi[9];
#pragma unroll
        for (int i = 0; i < 9; ++i) {
            si[i] = __shfl(s, i, 32);
            vi[i] = __shfl(v, i, 32);
        }
        float mx = NEG, vbest = NEG;
        int bp = 0;
#pragma unroll
        for (int i = 0; i < 9; ++i) {
            const float cs = si[i] + trcol[i];
            mx = fmaxf(mx, cs);
            const float cv = vi[i] + trcol[i];
            if (cv > vbest) { vbest = cv; bp = i; }
        }
        float acc = 0.0f;
#pragma unroll
        for (int i = 0; i < 9; ++i) acc += __expf(si[i] + trcol[i] - mx);
        const float ns = mx + __logf(acc) + em;
        const float nv = vbest + em;
        if (mt) { s = ns; v = nv; }

        if (valid) bp_lds[t * 9 + j] = (unsigned char)bp;
        if (j == 0) {
            mask_lds[t] = (unsigned char)mt;
            const int lab = labels[base + t];
            const int tag = (lab == -100) ? 0 : lab;
            if (mt) num += logits[(base + t) * 9 + tag] + trans[tag_prev * 9 + tag];
            tag_prev = tag;
            len += mt;
        }
    }

    // terminal reductions (shuffles executed by all lanes)
    const float send = valid ? (s + end_t[j]) : NEG;
    const float vend = valid ? (v + end_t[j]) : NEG;
    float se[9], ve[9];
#pragma unroll
    for (int i = 0; i < 9; ++i) {
        se[i] = __shfl(send, i, 32);
        ve[i] = __shfl(vend, i, 32);
    }
    float mx = NEG, vbest = NEG;
    int best_last = 0;
#pragma unroll
    for (int i = 0; i < 9; ++i) {
        mx = fmaxf(mx, se[i]);
        if (ve[i] > vbest) { vbest = ve[i]; best_last = i; }
    }
    float acc = 0.0f;
#pragma unroll
    for (int i = 0; i < 9; ++i) acc += __expf(se[i] - mx);
    const float den = mx + __logf(acc);

    __syncthreads();   // bp_lds / mask_lds visible

    if (j == 0) {
        const int lab = labels[base + (len - 1)];
        const int ltag = (lab == -100) ? 0 : lab;
        num += end_t[ltag];
        numden[b] = num;
        numden[64 + b] = den;
        // Viterbi backtrack in LDS
        int cur = best_last;
        tags_lds[511] = (short)cur;
        for (int t = 511; t >= 1; --t) {
            if (mask_lds[t]) cur = (int)bp_lds[t * 9 + cur];
            tags_lds[t - 1] = (short)cur;
        }
    }
    __syncthreads();

    // write fake_logits = one_hot(tags) * maskf
    for (int idx = j; idx < 512 * 9; idx += 32) {
        const int t = idx / 9;
        const int l = idx - t * 9;
        const float val = (mask_lds[t] && l == (int)tags_lds[t]) ? 1.0f : 0.0f;
        out_fake[(base + t) * 9 + l] = val;
    }
}

// ---------------------------------------------------------------------------
// Kernel 3: loss = -mean(num - den)
// ---------------------------------------------------------------------------
__global__ __launch_bounds__(64) void loss_kernel(const float* __restrict__ numden,
                                                  float* __restrict__ out)
{
    __shared__ float sdata[64];
    const int t = threadIdx.x;
    sdata[t] = numden[t] - numden[64 + t];
    __syncthreads();
    for (int off = 32; off > 0; off >>= 1) {
        if (t < off) sdata[t] += sdata[t + off];
        __syncthreads();
    }
    if (t == 0) out[0] = -sdata[0] * (1.0f / 64.0f);
}

// ---------------------------------------------------------------------------
extern "C" void kernel_launch(void* const* d_in, const int* in_sizes, int n_in,
                              void* d_out, int out_size, void* d_ws, size_t ws_size,
                              hipStream_t stream) {
    const float* hidden  = (const float*)d_in[0];
    const int*   mask    = (const int*)  d_in[1];
    const int*   labels  = (const int*)  d_in[2];
    const float* W       = (const float*)d_in[3];
    const float* bvec    = (const float*)d_in[4];
    const float* start_t = (const float*)d_in[5];
    const float* end_t   = (const float*)d_in[6];
    const float* trans   = (const float*)d_in[7];

    float* out    = (float*)d_out;           // [0]=loss, [1..]=fake_logits
    float* logits = (float*)d_ws;            // 32768*9 f32
    float* numden = logits + 32768 * 9;      // 128 f32

    logits_wmma_kernel<<<256, 256, 0, stream>>>(hidden, W, bvec, logits);
    crf_kernel<<<64, 32, 0, stream>>>(logits, mask, labels, start_t, end_t,
                                      trans, numden, out + 1);
    loss_kernel<<<1, 64, 0, stream>>>(numden, out);
}